// SpatialProcessor_46102178955809
// MI455X (gfx1250) — compile-verified
//
#include <hip/hip_runtime.h>
#include <hip/hip_bf16.h>
#include <math.h>
#include <limits.h>

// ---------------- problem constants (match reference) ----------------
#define N_NODES 10000
#define F_IN    256
#define UNITS   64
#define HEADS1  4
#define EMB     16
#define TOPK    20
#define NEG_SLOPE 0.2f
#define EDGES   (N_NODES * TOPK + N_NODES)   // 210000 (topk edges + self loops)

typedef __attribute__((ext_vector_type(2))) float v2f;
typedef __attribute__((ext_vector_type(8))) float v8f;

// lexicographic (value desc, index asc) comparison -> matches jax top_k tie-break
__device__ __forceinline__ bool pair_better(float va, int ia, float vb, int ib) {
  return (va > vb) || (va == vb && ia < ib);
}

// CAS-based float atomic max (portable, order-independent => deterministic)
__device__ __forceinline__ void atomic_max_f32(float* addr, float val) {
  unsigned int* ua = (unsigned int*)addr;
  unsigned int old = __float_as_uint(*addr);
  while (__uint_as_float(old) < val) {
    unsigned int assumed = old;
    old = atomicCAS(ua, assumed, __float_as_uint(val));
    if (old == assumed) break;
  }
}

// ---------------- 1) row-normalize node embeddings ----------------
__global__ void normalize_kernel(const float* __restrict__ emb, float* __restrict__ nrm) {
  int n = blockIdx.x * blockDim.x + threadIdx.x;
  if (n >= N_NODES) return;
  const float* e = emb + (size_t)n * EMB;
  float ss = 0.f;
#pragma unroll
  for (int k = 0; k < EMB; ++k) ss += e[k] * e[k];
  float inv = 1.0f / (sqrtf(ss) + 1e-12f);
  float* o = nrm + (size_t)n * EMB;
#pragma unroll
  for (int k = 0; k < EMB; ++k) o[k] = e[k] * inv;
}

// ---------------- 2) streaming top-K cosine neighbors ----------------
// one 256-thread workgroup per row; sim row never materialized to memory
#define TK_TPB 256
__global__ __launch_bounds__(TK_TPB) void topk_kernel(const float* __restrict__ nrm,
                                                      int* __restrict__ topk) {
  const int row = blockIdx.x;
  const int tid = threadIdx.x;

  __shared__ float rowv[EMB];
  if (tid < EMB) rowv[tid] = nrm[(size_t)row * EMB + tid];
  __syncthreads();

  float r0x = rowv[0],  r0y = rowv[1],  r0z = rowv[2],  r0w = rowv[3];
  float r1x = rowv[4],  r1y = rowv[5],  r1z = rowv[6],  r1w = rowv[7];
  float r2x = rowv[8],  r2y = rowv[9],  r2z = rowv[10], r2w = rowv[11];
  float r3x = rowv[12], r3y = rowv[13], r3z = rowv[14], r3w = rowv[15];

  float bestv[TOPK];
  int   besti[TOPK];
#pragma unroll
  for (int t = 0; t < TOPK; ++t) { bestv[t] = -INFINITY; besti[t] = INT_MAX; }

  for (int j = tid; j < N_NODES; j += TK_TPB) {
    const float4* np = (const float4*)(nrm + (size_t)j * EMB);
    float4 x0 = np[0], x1 = np[1], x2 = np[2], x3 = np[3];
    float s = x0.x*r0x + x0.y*r0y + x0.z*r0z + x0.w*r0w
            + x1.x*r1x + x1.y*r1y + x1.z*r1z + x1.w*r1w
            + x2.x*r2x + x2.y*r2y + x2.z*r2z + x2.w*r2w
            + x3.x*r3x + x3.y*r3y + x3.z*r3z + x3.w*r3w;
    if (pair_better(s, j, bestv[TOPK-1], besti[TOPK-1])) {
      // register-resident bubble insert (constant indexing only)
      float cv = s; int ci = j;
#pragma unroll
      for (int t = 0; t < TOPK; ++t) {
        if (pair_better(cv, ci, bestv[t], besti[t])) {
          float tv = bestv[t]; int ti = besti[t];
          bestv[t] = cv; besti[t] = ci;
          cv = tv; ci = ti;
        }
      }
    }
  }

  // candidate pool + 20 rounds of block-wide argmax
  __shared__ float cv[TK_TPB * TOPK];
  __shared__ int   ci[TK_TPB * TOPK];
  __shared__ float rv[TK_TPB];
  __shared__ int   ri[TK_TPB];
  __shared__ int   rp[TK_TPB];
#pragma unroll
  for (int t = 0; t < TOPK; ++t) { cv[tid*TOPK + t] = bestv[t]; ci[tid*TOPK + t] = besti[t]; }
  __syncthreads();

  for (int r = 0; r < TOPK; ++r) {
    int   bp = tid * TOPK;
    float bv = cv[bp];
    int   bi = ci[bp];
#pragma unroll
    for (int t = 1; t < TOPK; ++t) {
      int p = tid * TOPK + t;
      if (pair_better(cv[p], ci[p], bv, bi)) { bv = cv[p]; bi = ci[p]; bp = p; }
    }
    rv[tid] = bv; ri[tid] = bi; rp[tid] = bp;
    __syncthreads();
    for (int sft = TK_TPB / 2; sft > 0; sft >>= 1) {
      if (tid < sft) {
        if (pair_better(rv[tid + sft], ri[tid + sft], rv[tid], ri[tid])) {
          rv[tid] = rv[tid + sft]; ri[tid] = ri[tid + sft]; rp[tid] = rp[tid + sft];
        }
      }
      __syncthreads();
    }
    if (tid == 0) {
      topk[(size_t)row * TOPK + r] = ri[0];
      cv[rp[0]] = -INFINITY; ci[rp[0]] = INT_MAX;
    }
    __syncthreads();
  }
}

// ---------------- 3) fp32 WMMA GEMM: H = X(MxKd) @ W(KdxNcols) ----------------
// one wave computes a 16x64 output slab: 4 x V_WMMA_F32_16X16X4_F32 per K-step,
// A fragment loaded once and reused by all four column sub-tiles.
// Requires Ncols % 64 == 0, M % 16 == 0, Kd % 4 == 0.
__global__ void wmma_gemm_f32(const float* __restrict__ X, const float* __restrict__ W,
                              float* __restrict__ H, int M, int Kd, int Ncols) {
  const int wavesPerBlock = blockDim.x >> 5;
  const int waveId = blockIdx.x * wavesPerBlock + (threadIdx.x >> 5);
  const int lane = threadIdx.x & 31;
  const int tilesM  = M >> 4;
  const int tilesN4 = Ncols >> 6;                 // 64-wide column groups
  if (waveId >= tilesM * tilesN4) return;         // wave-uniform: EXEC stays all-1s

  const int row0 = (waveId / tilesN4) << 4;
  const int col0 = (waveId % tilesN4) << 6;
  const int half = lane >> 4;        // 0: K pair {0,1}, 1: K pair {2,3}
  const int l16  = lane & 15;

  v8f c0 = {}, c1 = {}, c2 = {}, c3 = {};
  const float* arow = X + (size_t)(row0 + l16) * Kd + 2 * half;   // A base for this lane
  for (int k = 0; k < Kd; k += 4) {
    // A fragment (16x4, §7.12.2): lane holds X[row0+l16][k + 2*half + {0,1}]
    v2f a; a.x = arow[k]; a.y = arow[k + 1];
    // B fragments (4x16 each): lane holds W[k + 2*half + {0,1}][col0 + 16*sub + l16]
    const float* bp = W + (size_t)(k + 2 * half) * Ncols + col0 + l16;
    v2f b0, b1, b2, b3;
    b0.x = bp[0];  b0.y = bp[Ncols];
    b1.x = bp[16]; b1.y = bp[16 + Ncols];
    b2.x = bp[32]; b2.y = bp[32 + Ncols];
    b3.x = bp[48]; b3.y = bp[48 + Ncols];
    c0 = __builtin_amdgcn_wmma_f32_16x16x4_f32(false, a, false, b0, (short)0, c0, false, false);
    c1 = __builtin_amdgcn_wmma_f32_16x16x4_f32(false, a, false, b1, (short)0, c1, false, false);
    c2 = __builtin_amdgcn_wmma_f32_16x16x4_f32(false, a, false, b2, (short)0, c2, false, false);
    c3 = __builtin_amdgcn_wmma_f32_16x16x4_f32(false, a, false, b3, (short)0, c3, false, false);
  }
  // C/D layout: VGPR r -> row = row0 + r + 8*half, col = col0 + 16*sub + l16
#pragma unroll
  for (int r = 0; r < 8; ++r) {
    float* o = H + (size_t)(row0 + r + 8 * half) * Ncols + col0 + l16;
    o[0]  = c0[r];
    o[16] = c1[r];
    o[32] = c2[r];
    o[48] = c3[r];
  }
}

// ---------------- 4) per-node attention scores ----------------
__global__ void score_kernel(const float* __restrict__ H, const float* __restrict__ a_s,
                             const float* __restrict__ a_d, float* __restrict__ s_score,
                             float* __restrict__ d_score, int heads, int units) {
  int idx = blockIdx.x * blockDim.x + threadIdx.x;   // n*heads + h
  if (idx >= N_NODES * heads) return;
  int n = idx / heads, h = idx % heads;
  const float* hv = H + (size_t)n * heads * units + (size_t)h * units;
  float ss = 0.f, ds = 0.f;
  for (int u = 0; u < units; ++u) {
    float v = hv[u];
    ss += v * a_s[h * units + u];
    ds += v * a_d[h * units + u];
  }
  s_score[idx] = ss;
  d_score[idx] = ds;
}

// ---------------- helpers ----------------
__global__ void fill_kernel(float* __restrict__ p, float v, int n) {
  int i = blockIdx.x * blockDim.x + threadIdx.x;
  if (i < n) p[i] = v;
}

__device__ __forceinline__ void edge_nodes(int e, const int* __restrict__ topk, int& i, int& j) {
  if (e < N_NODES * TOPK) { i = e / TOPK; j = topk[e]; }
  else                    { i = e - N_NODES * TOPK; j = i; }
}

__device__ __forceinline__ float leaky(float x) { return x > 0.f ? x : NEG_SLOPE * x; }

// ---------------- 5) segment max over in-edges ----------------
__global__ void edge_max_kernel(const float* __restrict__ s_score, const float* __restrict__ d_score,
                                const int* __restrict__ topk, float* __restrict__ m, int heads) {
  int e = blockIdx.x * blockDim.x + threadIdx.x;
  if (e >= EDGES) return;
  int i, j; edge_nodes(e, topk, i, j);
  for (int h = 0; h < heads; ++h) {
    float lg = leaky(s_score[i * heads + h] + d_score[j * heads + h]);
    atomic_max_f32(&m[j * heads + h], lg);
  }
}

// ---------------- 6) exp + segment sum ----------------
__global__ void edge_expsum_kernel(const float* __restrict__ s_score, const float* __restrict__ d_score,
                                   const int* __restrict__ topk, const float* __restrict__ m,
                                   float* __restrict__ edge_e, float* __restrict__ denom, int heads) {
  int e = blockIdx.x * blockDim.x + threadIdx.x;
  if (e >= EDGES) return;
  int i, j; edge_nodes(e, topk, i, j);
  for (int h = 0; h < heads; ++h) {
    float lg = leaky(s_score[i * heads + h] + d_score[j * heads + h]);
    float ev = expf(lg - m[j * heads + h]);
    edge_e[(size_t)e * heads + h] = ev;
    atomicAdd(&denom[j * heads + h], ev);
  }
}

// ---------------- 7) weighted message scatter ----------------
// thread per (edge, unit), loop over heads
__global__ void message_kernel(const float* __restrict__ H, const int* __restrict__ topk,
                               const float* __restrict__ edge_e, const float* __restrict__ denom,
                               float* __restrict__ accum, int heads, int units) {
  long long t = (long long)blockIdx.x * blockDim.x + threadIdx.x;
  if (t >= (long long)EDGES * units) return;
  int e = (int)(t / units);
  int u = (int)(t % units);
  int i, j; edge_nodes(e, topk, i, j);
  for (int h = 0; h < heads; ++h) {
    float alpha = edge_e[(size_t)e * heads + h] / (denom[j * heads + h] + 1e-9f);
    float msg = H[((size_t)i * heads + h) * units + u] * alpha;
    atomicAdd(&accum[((size_t)j * heads + h) * units + u], msg);
  }
}

// ---------------- 8) x = relu(x + b) in place ----------------
__global__ void bias_relu_kernel(float* __restrict__ x, const float* __restrict__ b,
                                 int total, int cols) {
  int idx = blockIdx.x * blockDim.x + threadIdx.x;
  if (idx >= total) return;
  float v = x[idx] + b[idx % cols];
  x[idx] = v > 0.f ? v : 0.f;
}

// ---------------- 9) seed output with bias b2 ----------------
__global__ void init_out_kernel(float* __restrict__ out, const float* __restrict__ b2) {
  int idx = blockIdx.x * blockDim.x + threadIdx.x;
  if (idx >= N_NODES * UNITS) return;
  out[idx] = b2[idx % UNITS];
}

// =====================================================================
extern "C" void kernel_launch(void* const* d_in, const int* in_sizes, int n_in,
                              void* d_out, int out_size, void* d_ws, size_t ws_size,
                              hipStream_t stream) {
  const float* inputs   = (const float*)d_in[0];
  const float* node_emb = (const float*)d_in[1];
  const float* W1       = (const float*)d_in[2];
  const float* a_src1   = (const float*)d_in[3];
  const float* a_dst1   = (const float*)d_in[4];
  const float* b1       = (const float*)d_in[5];
  const float* W2       = (const float*)d_in[6];
  const float* a_src2   = (const float*)d_in[7];
  const float* a_dst2   = (const float*)d_in[8];
  const float* b2       = (const float*)d_in[9];
  float* out = (float*)d_out;
  float* ws  = (float*)d_ws;

  // workspace layout (float slots), ~29.5 MB total -> L2 resident
  float* nrm     = ws;                    // 160000
  int*   topk    = (int*)(ws + 160000);   // 200000
  float* h1      = ws + 360000;           // 2560000  (N x 256)
  float* s1      = ws + 2920000;          // 40000
  float* d1      = ws + 2960000;          // 40000
  float* m1      = ws + 3000000;          // 40000
  float* den1    = ws + 3040000;          // 40000
  float* ee1     = ws + 3080000;          // 840000   (E x 4)
  float* acc1    = ws + 3920000;          // 2560000  (N x 256) -> becomes x1 after bias+relu
  float* h2      = ws + 6480000;          // 640000   (N x 64)
  float* s2      = ws + 7120000;          // 10000
  float* d2      = ws + 7130000;          // 10000
  float* m2      = ws + 7140000;          // 10000
  float* den2    = ws + 7150000;          // 10000
  float* ee2     = ws + 7160000;          // 210000
  (void)in_sizes; (void)n_in; (void)out_size; (void)ws_size;

  const int TPB = 256;
  auto blks = [](long long n, int tpb) { return (int)((n + tpb - 1) / tpb); };

  // ---- edge construction ----
  normalize_kernel<<<blks(N_NODES, TPB), TPB, 0, stream>>>(node_emb, nrm);
  topk_kernel<<<N_NODES, TK_TPB, 0, stream>>>(nrm, topk);

  // ---- GAT layer 1 ----
  // h1 = inputs @ W1 : 625 M-tiles x 4 column-groups = 2500 waves, 8 waves/block
  wmma_gemm_f32<<<blks(625 * 4, 8), 256, 0, stream>>>(inputs, W1, h1, N_NODES, F_IN, HEADS1 * UNITS);
  score_kernel<<<blks((long long)N_NODES * HEADS1, TPB), TPB, 0, stream>>>(
      h1, a_src1, a_dst1, s1, d1, HEADS1, UNITS);
  fill_kernel<<<blks(N_NODES * HEADS1, TPB), TPB, 0, stream>>>(m1, -INFINITY, N_NODES * HEADS1);
  fill_kernel<<<blks(N_NODES * HEADS1, TPB), TPB, 0, stream>>>(den1, 0.f, N_NODES * HEADS1);
  fill_kernel<<<blks(N_NODES * HEADS1 * UNITS, TPB), TPB, 0, stream>>>(acc1, 0.f, N_NODES * HEADS1 * UNITS);
  edge_max_kernel<<<blks(EDGES, TPB), TPB, 0, stream>>>(s1, d1, topk, m1, HEADS1);
  edge_expsum_kernel<<<blks(EDGES, TPB), TPB, 0, stream>>>(s1, d1, topk, m1, ee1, den1, HEADS1);
  message_kernel<<<blks((long long)EDGES * UNITS, TPB), TPB, 0, stream>>>(
      h1, topk, ee1, den1, acc1, HEADS1, UNITS);
  bias_relu_kernel<<<blks(N_NODES * HEADS1 * UNITS, TPB), TPB, 0, stream>>>(
      acc1, b1, N_NODES * HEADS1 * UNITS, HEADS1 * UNITS);

  // ---- GAT layer 2 ----
  // h2 = x1 @ W2 : 625 M-tiles x 1 column-group = 625 waves
  wmma_gemm_f32<<<blks(625, 8), 256, 0, stream>>>(acc1, W2, h2, N_NODES, HEADS1 * UNITS, UNITS);
  score_kernel<<<blks(N_NODES, TPB), TPB, 0, stream>>>(h2, a_src2, a_dst2, s2, d2, 1, UNITS);
  fill_kernel<<<blks(N_NODES, TPB), TPB, 0, stream>>>(m2, -INFINITY, N_NODES);
  fill_kernel<<<blks(N_NODES, TPB), TPB, 0, stream>>>(den2, 0.f, N_NODES);
  edge_max_kernel<<<blks(EDGES, TPB), TPB, 0, stream>>>(s2, d2, topk, m2, 1);
  edge_expsum_kernel<<<blks(EDGES, TPB), TPB, 0, stream>>>(s2, d2, topk, m2, ee2, den2, 1);
  init_out_kernel<<<blks(N_NODES * UNITS, TPB), TPB, 0, stream>>>(out, b2);
  message_kernel<<<blks((long long)EDGES * UNITS, TPB), TPB, 0, stream>>>(
      h2, topk, ee2, den2, out, 1, UNITS);
}